// Attention_BiRNN_Layer_v2_10960756539880
// MI455X (gfx1250) — compile-verified
//
#include <hip/hip_runtime.h>
#include <hip/hip_bf16.h>

#define B_ 64
#define S_ 512
#define F_ 768
#define NEG_INF_ (-1e30f)

typedef __attribute__((ext_vector_type(2))) float v2f;
typedef __attribute__((ext_vector_type(4))) float v4f;
typedef __attribute__((ext_vector_type(8))) float v8f;

// ---------------------------------------------------------------------------
// Kernel 1: temp[r] = dot(x[r, :], cv)  for r in [0, B*S)   (GEMV via WMMA)
// Block = 256 threads (8 wave32) -> 128 rows/block; grid = 256 blocks.
// Each wave owns a 16-row tile; K consumed 4 at a time by
// V_WMMA_F32_16X16X4_F32 with B = cv broadcast across all 16 columns.
// ---------------------------------------------------------------------------
#define K1_ROWS 128
#define K1_KC    64
#define K1_LDST  68   // padded LDS row stride (dwords): conflict-free A reads

__global__ void __launch_bounds__(256)
k1_matvec_wmma(const float* __restrict__ x, const float* __restrict__ cv,
               float* __restrict__ temp) {
  __shared__ float tile[K1_ROWS * K1_LDST];
  __shared__ float cvs[F_];
  const int tid   = threadIdx.x;
  const int lane  = tid & 31;
  const int wave  = tid >> 5;
  const int laneM = lane & 15;
  const int half  = lane >> 4;       // 0: K={0,1}, 1: K={2,3}
  const int koff  = half * 2;
  const int rowBase = blockIdx.x * K1_ROWS;

  for (int i = tid; i < F_; i += 256) cvs[i] = cv[i];

  v8f acc = {};
  for (int k0 = 0; k0 < F_; k0 += K1_KC) {
    // Stage 128x64 f32 tile, coalesced b128 loads (16 threads/row, 256B/row)
    #pragma unroll
    for (int i = 0; i < 8; ++i) {
      int idx = tid + i * 256;            // 0..2047
      int r   = idx >> 4;                 // 0..127
      int c4  = (idx & 15) << 2;          // 0,4,...,60
      v4f v = *(const v4f*)(x + (size_t)(rowBase + r) * F_ + k0 + c4);
      *(v4f*)(&tile[r * K1_LDST + c4]) = v;
    }
    __syncthreads();

    const int lrow = wave * 16 + laneM;
    const float* arow = &tile[lrow * K1_LDST + koff];
    const float* brow = &cvs[k0 + koff];
    #pragma unroll
    for (int kk = 0; kk < K1_KC; kk += 4) {
      v2f av = *(const v2f*)(arow + kk);   // A[M=laneM][K = koff, koff+1]
      v2f bv = *(const v2f*)(brow + kk);   // B[K][n] = cv[k] for all n
      acc = __builtin_amdgcn_wmma_f32_16x16x4_f32(
          /*neg_a=*/false, av, /*neg_b=*/false, bv,
          /*c_mod=*/(short)0, acc, /*reuse_a=*/false, /*reuse_b=*/false);
    }
    __syncthreads();
  }

  // C/D layout: VGPR j, lanes<16 -> M=j, lanes>=16 -> M=j+8; all N identical.
  if (laneM == 0) {
    int r = rowBase + wave * 16 + half * 8;
    #pragma unroll
    for (int j = 0; j < 8; ++j) temp[r + j] = acc[j];
  }
}

// ---------------------------------------------------------------------------
// Kernel 2: bi-LSTM(1,1) scans + masked softmax over S, one block per batch.
// ---------------------------------------------------------------------------
__device__ __forceinline__ float sigf(float x) {
  return 1.0f / (1.0f + __expf(-x));
}
__device__ __forceinline__ float tanh_fast(float x) {
  // robust for large |x|: e->inf gives 1, e->0 gives -1
  float e = __expf(2.0f * x);
  return 1.0f - 2.0f / (e + 1.0f);
}

__global__ void __launch_bounds__(128)
k2_lstm_softmax(const float* __restrict__ temp,
                const unsigned char* __restrict__ mask,
                const float* __restrict__ wif, const float* __restrict__ whf,
                const float* __restrict__ bif, const float* __restrict__ bhf,
                const float* __restrict__ wib, const float* __restrict__ whb,
                const float* __restrict__ bib, const float* __restrict__ bhb,
                float* __restrict__ aOut, float* __restrict__ aWs) {
  __shared__ float h1[S_];
  __shared__ float h2[S_];
  __shared__ float red[128];
  const int b   = blockIdx.x;
  const int tid = threadIdx.x;
  const float* xb = temp + b * S_;

  if (tid == 0) {                       // forward scan
    float w0 = wif[0], w1 = wif[1], w2 = wif[2], w3 = wif[3];
    float u0 = whf[0], u1 = whf[1], u2 = whf[2], u3 = whf[3];
    float c0 = bif[0] + bhf[0], c1 = bif[1] + bhf[1];
    float c2 = bif[2] + bhf[2], c3 = bif[3] + bhf[3];
    float h = 0.f, c = 0.f;
    for (int s = 0; s < S_; ++s) {
      float xt = xb[s];
      float gi = fmaf(xt, w0, fmaf(h, u0, c0));
      float gf = fmaf(xt, w1, fmaf(h, u1, c1));
      float gg = fmaf(xt, w2, fmaf(h, u2, c2));
      float go = fmaf(xt, w3, fmaf(h, u3, c3));
      float i_ = sigf(gi), f_ = sigf(gf), g_ = tanh_fast(gg), o_ = sigf(go);
      c = fmaf(f_, c, i_ * g_);
      h = o_ * tanh_fast(c);
      h1[s] = h;
    }
  } else if (tid == 32) {               // backward scan (flip-scan-flip)
    float w0 = wib[0], w1 = wib[1], w2 = wib[2], w3 = wib[3];
    float u0 = whb[0], u1 = whb[1], u2 = whb[2], u3 = whb[3];
    float c0 = bib[0] + bhb[0], c1 = bib[1] + bhb[1];
    float c2 = bib[2] + bhb[2], c3 = bib[3] + bhb[3];
    float h = 0.f, c = 0.f;
    for (int s = S_ - 1; s >= 0; --s) {
      float xt = xb[s];
      float gi = fmaf(xt, w0, fmaf(h, u0, c0));
      float gf = fmaf(xt, w1, fmaf(h, u1, c1));
      float gg = fmaf(xt, w2, fmaf(h, u2, c2));
      float go = fmaf(xt, w3, fmaf(h, u3, c3));
      float i_ = sigf(gi), f_ = sigf(gf), g_ = tanh_fast(gg), o_ = sigf(go);
      c = fmaf(f_, c, i_ * g_);
      h = o_ * tanh_fast(c);
      h2[s] = h;
    }
  }
  __syncthreads();

  // masked values + block max
  float mx = -__builtin_inff();
  #pragma unroll
  for (int i = 0; i < 4; ++i) {
    int s = tid + i * 128;
    float v = h1[s] + h2[s];
    v = mask[b * S_ + s] ? v : NEG_INF_;
    h1[s] = v;
    mx = fmaxf(mx, v);
  }
  red[tid] = mx; __syncthreads();
  for (int o = 64; o > 0; o >>= 1) {
    if (tid < o) red[tid] = fmaxf(red[tid], red[tid + o]);
    __syncthreads();
  }
  mx = red[0];
  __syncthreads();

  // sum of exp
  float sum = 0.f;
  #pragma unroll
  for (int i = 0; i < 4; ++i) {
    int s = tid + i * 128;
    float e = __expf(h1[s] - mx);
    h1[s] = e;
    sum += e;
  }
  red[tid] = sum; __syncthreads();
  for (int o = 64; o > 0; o >>= 1) {
    if (tid < o) red[tid] += red[tid + o];
    __syncthreads();
  }
  float inv = 1.0f / red[0];

  #pragma unroll
  for (int i = 0; i < 4; ++i) {
    int s = tid + i * 128;
    float a = h1[s] * inv;
    aOut[b * S_ + s] = a;
    aWs[b * S_ + s]  = a;
  }
}

// ---------------------------------------------------------------------------
// Kernel 3: out[b,f] = min_s( x[b,s,f] * a[b,s] ).  One thread per (b,f).
// grid = (F/256, B); a[b,:] cached in LDS; x columns coalesced across f.
// ---------------------------------------------------------------------------
__global__ void __launch_bounds__(256)
k3_minpool(const float* __restrict__ x, const float* __restrict__ aWs,
           float* __restrict__ out) {
  __shared__ float aLds[S_];
  const int tid = threadIdx.x;
  const int b   = blockIdx.y;
  const int f   = blockIdx.x * 256 + tid;
  aLds[tid]       = aWs[b * S_ + tid];
  aLds[tid + 256] = aWs[b * S_ + tid + 256];
  __syncthreads();

  const float* xp = x + (size_t)b * S_ * F_ + f;
  float m = __builtin_inff();
  #pragma unroll 1
  for (int s = 0; s < S_; s += 4) {
    if ((tid & 15) == 0)  // one prefetch per 64B line, 24 rows ahead
      __builtin_prefetch(xp + (size_t)(s + 24) * F_, 0, 1);
    float v0 = xp[(size_t)(s + 0) * F_] * aLds[s + 0];
    float v1 = xp[(size_t)(s + 1) * F_] * aLds[s + 1];
    float v2 = xp[(size_t)(s + 2) * F_] * aLds[s + 2];
    float v3 = xp[(size_t)(s + 3) * F_] * aLds[s + 3];
    m = fminf(m, fminf(fminf(v0, v1), fminf(v2, v3)));
  }
  out[b * F_ + f] = m;
}

// ---------------------------------------------------------------------------
extern "C" void kernel_launch(void* const* d_in, const int* in_sizes, int n_in,
                              void* d_out, int out_size, void* d_ws, size_t ws_size,
                              hipStream_t stream) {
  const float* x  = (const float*)d_in[0];
  const unsigned char* mask = (const unsigned char*)d_in[1];  // jax bool = 1 byte
  const float* cv  = (const float*)d_in[2];
  const float* wif = (const float*)d_in[3];
  const float* whf = (const float*)d_in[4];
  const float* bif = (const float*)d_in[5];
  const float* bhf = (const float*)d_in[6];
  const float* wib = (const float*)d_in[7];
  const float* whb = (const float*)d_in[8];
  const float* bib = (const float*)d_in[9];
  const float* bhb = (const float*)d_in[10];

  float* out  = (float*)d_out;          // [B*F] min-pool, then [B*S] attention
  float* temp = (float*)d_ws;           // [B*S] context projection
  float* aWs  = temp + B_ * S_;         // [B*S] attention weights (scratch)

  k1_matvec_wmma<<<dim3((B_ * S_) / K1_ROWS), dim3(256), 0, stream>>>(x, cv, temp);
  k2_lstm_softmax<<<dim3(B_), dim3(128), 0, stream>>>(
      temp, mask, wif, whf, bif, bhf, wib, whb, bib, bhb, out + B_ * F_, aWs);
  k3_minpool<<<dim3(F_ / 256, B_), dim3(256), 0, stream>>>(x, aWs, out);
}